// Model_21981642621250
// MI455X (gfx1250) — compile-verified
//
#include <hip/hip_runtime.h>
#include <math.h>

// ---- problem dims (match reference) ----
#define Bb  2
#define Nn  128
#define Hh  768
#define Kk  64
#define H2d 256
#define H3d 128
#define H4d 32
#define NCd 7
#define FC1d 1000
#define FC2d 100

typedef float v2f __attribute__((ext_vector_type(2)));
typedef float v8f __attribute__((ext_vector_type(8)));

// =====================================================================
// Generic f32 WMMA GEMM:  C = epilogue(A * B(^T) + bias + resid)
// One wave per 16x16 output tile, K-loop in steps of 4 using
// V_WMMA_F32_16X16X4_F32. Requires Kd % 4 == 0.
// Wave-uniform fast path for fully in-bounds tiles (raw b64 loads feed
// WMMA directly); guarded clamp+zero path only for ragged edge tiles.
// PREFETCH streams B ahead via global_prefetch (fc1_w = 196.6MB is the
// bandwidth floor of the whole net at 23.3 TB/s).
// =====================================================================
template <int TRANSB, int PREFETCH>
__global__ void gemm_wmma_f32(const float* __restrict__ A,
                              const float* __restrict__ B,
                              float* __restrict__ C,
                              int M, int N, int Kd,
                              int lda, int ldb, int ldc,
                              long sA, long sB, long sC,
                              const float* __restrict__ bias,
                              const float* __restrict__ resid, long sR,
                              int epilogue)  // 0 none, 1 relu, 2 gelu(erf)
{
    const int lane = threadIdx.x & 31;
    const int wave = threadIdx.x >> 5;
    const int tilesN = (N + 15) >> 4;
    const int tilesM = (M + 15) >> 4;
    const int tile = blockIdx.x * (blockDim.x >> 5) + wave;
    if (tile >= tilesM * tilesN) return;          // wave-uniform exit
    const int m0 = (tile / tilesN) << 4;
    const int n0 = (tile % tilesN) << 4;
    const int batch = blockIdx.y;
    A += (long)batch * sA;
    B += (long)batch * sB;
    C += (long)batch * sC;

    // A-matrix 16x4 layout: lane holds row (lane&15), K pair (lane>>4)*2 + {0,1}
    const int rowA  = m0 + (lane & 15);
    const int rA    = rowA < M ? rowA : (M - 1);
    const float aok = (rowA < M) ? 1.0f : 0.0f;
    // B-matrix 4x16 layout (mirrors A): lane holds col (lane&15), same K pair
    const int colB  = n0 + (lane & 15);
    const int cB    = colB < N ? colB : (N - 1);
    const float bok = (colB < N) ? 1.0f : 0.0f;
    const int koff  = (lane >> 4) << 1;           // 0 or 2

    const float* pA = A + (long)rA * lda + koff;
    const float* pB = TRANSB ? (B + (long)cB * ldb + koff)
                             : (B + (long)koff * ldb + cB);
    const long bstep = TRANSB ? 4 : (long)4 * ldb;

    const bool fullTile = (m0 + 16 <= M) && (n0 + 16 <= N);  // wave-uniform

    v8f acc = {};
    if (fullTile) {
#pragma unroll 4
        for (int k = 0; k < Kd; k += 4) {
            v2f a, b;
            a.x = pA[0];
            a.y = pA[1];
            if (TRANSB) { b.x = pB[0]; b.y = pB[1]; }
            else        { b.x = pB[0]; b.y = pB[ldb]; }
            if (PREFETCH) __builtin_prefetch(pB + (long)64 * ldb, 0, 1);
            acc = __builtin_amdgcn_wmma_f32_16x16x4_f32(
                      false, a, false, b, (short)0, acc, false, false);
            pA += 4;
            pB += bstep;
        }
    } else {
#pragma unroll 4
        for (int k = 0; k < Kd; k += 4) {
            v2f a, b;
            a.x = aok * pA[0];
            a.y = aok * pA[1];
            if (TRANSB) { b.x = bok * pB[0]; b.y = bok * pB[1]; }
            else        { b.x = bok * pB[0]; b.y = bok * pB[ldb]; }
            if (PREFETCH) __builtin_prefetch(pB + (long)64 * ldb, 0, 1);
            acc = __builtin_amdgcn_wmma_f32_16x16x4_f32(
                      false, a, false, b, (short)0, acc, false, false);
            pA += 4;
            pB += bstep;
        }
    }

    // C/D layout: lane holds col (lane&15), rows m0 + (lane>=16?8:0) + v
    const int rbase = (lane >> 4) << 3;
    const int col   = n0 + (lane & 15);
    const float bv  = (bias && col < N) ? bias[col] : 0.0f;
    const float* R  = resid ? (resid + (long)batch * sR) : nullptr;
    if (fullTile) {
        float* pC = C + (long)(m0 + rbase) * ldc + col;
        const float* pR = R ? (R + (long)(m0 + rbase) * ldc + col) : nullptr;
#pragma unroll
        for (int v = 0; v < 8; ++v) {
            float x = acc[v] + bv;
            if (pR) x += pR[(long)v * ldc];
            if (epilogue == 1)      x = fmaxf(x, 0.0f);
            else if (epilogue == 2) x = 0.5f * x * (1.0f + erff(x * 0.7071067811865475f));
            pC[(long)v * ldc] = x;
        }
    } else if (col < N) {
#pragma unroll
        for (int v = 0; v < 8; ++v) {
            const int row = m0 + rbase + v;
            if (row < M) {
                float x = acc[v] + bv;
                if (R) x += R[(long)row * ldc + col];
                if (epilogue == 1)      x = fmaxf(x, 0.0f);
                else if (epilogue == 2) x = 0.5f * x * (1.0f + erff(x * 0.7071067811865475f));
                C[(long)row * ldc + col] = x;
            }
        }
    }
}

// =====================================================================
// Row squared-norms for encoder rows (B*N of them) and landmarks (K).
// One 256-thread block per row of length H.
// =====================================================================
__global__ void norms_kernel(const float* __restrict__ enc,
                             const float* __restrict__ lmk,
                             float* __restrict__ en2,
                             float* __restrict__ ln2)
{
    __shared__ float red[256];
    const int row = blockIdx.x;
    const int t   = threadIdx.x;
    const float* src = (row < Bb * Nn) ? (enc + (long)row * Hh)
                                       : (lmk + (long)(row - Bb * Nn) * Hh);
    float s = 0.0f;
    for (int h = t; h < Hh; h += 256) { float v = src[h]; s += v * v; }
    red[t] = s; __syncthreads();
    for (int st = 128; st > 0; st >>= 1) {
        if (t < st) red[t] += red[t + st];
        __syncthreads();
    }
    if (t == 0) {
        if (row < Bb * Nn) en2[row] = red[0];
        else               ln2[row - Bb * Nn] = red[0];
    }
}

// =====================================================================
// dist -> min/max normalize over K -> gaussian sim * mask. One block per
// (b,n), 64 threads (one per cluster).
// =====================================================================
__global__ void sim_kernel(const float* __restrict__ dot,
                           const float* __restrict__ en2,
                           const float* __restrict__ ln2,
                           const float* __restrict__ mask,
                           const float* __restrict__ dparam,
                           float* __restrict__ P)
{
    __shared__ float smn[64], smx[64];
    const int bn = blockIdx.x;        // 0..B*N-1
    const int k  = threadIdx.x;       // 0..63
    float d2 = en2[bn] + ln2[k] - 2.0f * dot[(long)bn * Kk + k];
    float dist = sqrtf(fmaxf(d2, 0.0f));
    smn[k] = dist; smx[k] = dist; __syncthreads();
    for (int s = 32; s > 0; s >>= 1) {
        if (k < s) {
            smn[k] = fminf(smn[k], smn[k + s]);
            smx[k] = fmaxf(smx[k], smx[k + s]);
        }
        __syncthreads();
    }
    const float mn = smn[0], mx = smx[0];
    float nd = (dist - mn) / (mx - mn + 1e-6f);
    float dv = dparam[0];
    float sim = expf(-(nd * nd) / (2.0f * dv * dv)) * mask[bn];
    P[(long)bn * Kk + k] = sim;
}

// =====================================================================
// Normalize P over the token dim and write transposed copy PT[b,k,n].
// One block per (b,k), 128 threads (one per token).
// =====================================================================
__global__ void pnorm_kernel(float* __restrict__ P, float* __restrict__ PT)
{
    __shared__ float red[128];
    const int bk = blockIdx.x;            // b*K + k
    const int b  = bk / Kk, k = bk % Kk;
    const int n  = threadIdx.x;
    const long idx = ((long)b * Nn + n) * Kk + k;
    float v = P[idx];
    red[n] = v; __syncthreads();
    for (int s = 64; s > 0; s >>= 1) {
        if (n < s) red[n] += red[n + s];
        __syncthreads();
    }
    float p = v / (red[0] + 1e-6f);
    P[idx] = p;
    PT[((long)b * Kk + k) * Nn + n] = p;
}

// Wd[i,j] = exp(-alpha * |i-j|)
__global__ void wd_kernel(const float* __restrict__ alpha, float* __restrict__ Wd)
{
    const int idx = blockIdx.x * blockDim.x + threadIdx.x;
    if (idx >= Nn * Nn) return;
    const int i = idx >> 7, j = idx & (Nn - 1);
    Wd[idx] = expf(-alpha[0] * fabsf((float)(i - j)));
}

// =====================================================================
// G[b,i,j] = sum_{k,l} (exp(beta*p_ik*p_jl*Wd_kl) - 1); adjacency = (G!=0)|diag.
// Dominant kernel: B*K*K blocks x 16384 exp terms. Each term >= 0 so the
// !=0 test is reduction-order independent. Uses hardware v_exp_f32.
// =====================================================================
__global__ void graph_kernel(const float* __restrict__ PT,
                             const float* __restrict__ Wd,
                             const float* __restrict__ beta,
                             float* __restrict__ ADJ)
{
    __shared__ float pi[Nn], pj[Nn], red[256];
    const int b = blockIdx.y;
    const int i = blockIdx.x >> 6;
    const int j = blockIdx.x & (Kk - 1);
    const int t = threadIdx.x;
    if (t < Nn)       pi[t]      = PT[((long)b * Kk + i) * Nn + t];
    else              pj[t - Nn] = PT[((long)b * Kk + j) * Nn + (t - Nn)];
    __syncthreads();

    const float bet = beta[0];
    const int l = t & (Nn - 1);
    const float pjl = pj[l];
    float acc = 0.0f;
#pragma unroll 4
    for (int k = (t >> 7); k < Nn; k += 2) {
        float x = bet * pi[k] * pjl * Wd[k * Nn + l];
        acc += __expf(x) - 1.0f;          // each term >= 0
    }
    red[t] = acc; __syncthreads();
    for (int s = 128; s > 0; s >>= 1) {
        if (t < s) red[t] += red[t + s];
        __syncthreads();
    }
    if (t == 0) {
        float G = red[0];
        ADJ[((long)b * Kk + i) * Kk + j] = (i == j || G != 0.0f) ? 1.0f : 0.0f;
    }
}

// Symmetric normalization of binary adjacency, in place. One block per batch.
__global__ void nadj_kernel(float* __restrict__ ADJ)
{
    __shared__ float As[Kk * Kk];
    __shared__ float dinv[Kk];
    const int b = blockIdx.x;
    const int t = threadIdx.x;
    for (int i = t; i < Kk * Kk; i += 256) As[i] = ADJ[(long)b * Kk * Kk + i];
    __syncthreads();
    if (t < Kk) {
        float deg = 0.0f;
        for (int j = 0; j < Kk; ++j) deg += As[t * Kk + j];
        dinv[t] = (deg > 0.0f) ? (1.0f / sqrtf(deg)) : 0.0f;
    }
    __syncthreads();
    for (int i = t; i < Kk * Kk; i += 256) {
        const int r = i >> 6, c = i & (Kk - 1);
        ADJ[(long)b * Kk * Kk + i] = dinv[r] * As[i] * dinv[c];
    }
}

// ---------------------------------------------------------------------
static inline void launch_gemm(hipStream_t s,
                               const float* A, const float* B, float* C,
                               int M, int N, int Kd,
                               int lda, int ldb, int ldc,
                               long sA, long sB, long sC, int batches,
                               int transB, int prefetch,
                               const float* bias, const float* resid, long sR,
                               int ep)
{
    const int tiles = ((M + 15) / 16) * ((N + 15) / 16);
    const int wpb = 4;  // waves per block
    dim3 grid((tiles + wpb - 1) / wpb, batches);
    if (transB)
        gemm_wmma_f32<1, 0><<<grid, wpb * 32, 0, s>>>(A, B, C, M, N, Kd,
                lda, ldb, ldc, sA, sB, sC, bias, resid, sR, ep);
    else if (prefetch)
        gemm_wmma_f32<0, 1><<<grid, wpb * 32, 0, s>>>(A, B, C, M, N, Kd,
                lda, ldb, ldc, sA, sB, sC, bias, resid, sR, ep);
    else
        gemm_wmma_f32<0, 0><<<grid, wpb * 32, 0, s>>>(A, B, C, M, N, Kd,
                lda, ldb, ldc, sA, sB, sC, bias, resid, sR, ep);
}

extern "C" void kernel_launch(void* const* d_in, const int* in_sizes, int n_in,
                              void* d_out, int out_size, void* d_ws, size_t ws_size,
                              hipStream_t stream)
{
    const float* enc    = (const float*)d_in[0];
    const float* mask   = (const float*)d_in[1];
    const float* lmk    = (const float*)d_in[2];
    const float* dparam = (const float*)d_in[3];
    const float* alpha  = (const float*)d_in[4];
    const float* beta   = (const float*)d_in[5];
    const float* gw1 = (const float*)d_in[6],  * gb1 = (const float*)d_in[7];
    const float* gw2 = (const float*)d_in[8],  * gb2 = (const float*)d_in[9];
    const float* gw3 = (const float*)d_in[10], * gb3 = (const float*)d_in[11];
    const float* aw  = (const float*)d_in[12], * ab  = (const float*)d_in[13];
    const float* f1w = (const float*)d_in[14], * f1b = (const float*)d_in[15];
    const float* f2w = (const float*)d_in[16], * f2b = (const float*)d_in[17];
    const float* f3w = (const float*)d_in[18], * f3b = (const float*)d_in[19];
    float* out = (float*)d_out;

    // workspace layout (floats)
    float* w = (float*)d_ws;
    float* DOT  = w;                 w += Bb * Nn * Kk;      // 16384
    float* EN2  = w;                 w += Bb * Nn;           // 256
    float* LN2  = w;                 w += Kk;                // 64
    float* P    = w;                 w += Bb * Nn * Kk;      // 16384
    float* PT   = w;                 w += Bb * Kk * Nn;      // 16384
    float* WD   = w;                 w += Nn * Nn;           // 16384
    float* ADJ  = w;                 w += Bb * Kk * Kk;      // 8192
    float* NF   = w;                 w += Bb * Kk * Hh;      // 98304
    float* Tb   = w;                 w += Bb * Kk * H2d;     // 32768 (layer scratch)
    float* H1   = w;                 w += Bb * Kk * H2d;     // 32768
    float* H2b  = w;                 w += Bb * Kk * H3d;     // 16384
    float* H3   = w;                 w += Bb * Kk * H4d;     // 4096
    float* OUTR = w;                 w += Bb * Kk * Hh;      // 98304
    float* X1   = w;                 w += Bb * FC1d;         // 2000
    float* X2   = w;                 w += Bb * FC2d;         // 200

    // 1) row norms
    norms_kernel<<<Bb * Nn + Kk, 256, 0, stream>>>(enc, lmk, EN2, LN2);

    // 2) dot = E[256,768] @ L^T  (WMMA, transB)
    launch_gemm(stream, enc, lmk, DOT, Bb * Nn, Kk, Hh, Hh, Hh, Kk,
                0, 0, 0, 1, /*transB=*/1, 0, nullptr, nullptr, 0, 0);

    // 3) distances -> gaussian similarity
    sim_kernel<<<Bb * Nn, Kk, 0, stream>>>(DOT, EN2, LN2, mask, dparam, P);

    // 4) normalize over tokens, build P^T
    pnorm_kernel<<<Bb * Kk, Nn, 0, stream>>>(P, PT);

    // 5) Wd
    wd_kernel<<<(Nn * Nn + 255) / 256, 256, 0, stream>>>(alpha, WD);

    // 6) graph construct + binarize (hot kernel)
    {
        dim3 grid(Kk * Kk, Bb);
        graph_kernel<<<grid, 256, 0, stream>>>(PT, WD, beta, ADJ);
    }

    // 7) symmetric normalization
    nadj_kernel<<<Bb, 256, 0, stream>>>(ADJ);

    // 8) node_feat[b] = PT[b] (64x128) @ E[b] (128x768)
    launch_gemm(stream, PT, enc, NF, Kk, Hh, Nn, Nn, Hh, Hh,
                (long)Kk * Nn, (long)Nn * Hh, (long)Kk * Hh, Bb, 0, 0,
                nullptr, nullptr, 0, 0);

    // 9) GCN layer 1: H1 = relu(nadj @ (NF @ w1) + b1)
    launch_gemm(stream, NF, gw1, Tb, Kk, H2d, Hh, Hh, H2d, H2d,
                (long)Kk * Hh, 0, (long)Kk * H2d, Bb, 0, 0, nullptr, nullptr, 0, 0);
    launch_gemm(stream, ADJ, Tb, H1, Kk, H2d, Kk, Kk, H2d, H2d,
                (long)Kk * Kk, (long)Kk * H2d, (long)Kk * H2d, Bb, 0, 0,
                gb1, nullptr, 0, 1);

    // 10) GCN layer 2: H2 = relu(nadj @ (H1 @ w2) + b2)
    launch_gemm(stream, H1, gw2, Tb, Kk, H3d, H2d, H2d, H3d, H3d,
                (long)Kk * H2d, 0, (long)Kk * H3d, Bb, 0, 0, nullptr, nullptr, 0, 0);
    launch_gemm(stream, ADJ, Tb, H2b, Kk, H3d, Kk, Kk, H3d, H3d,
                (long)Kk * Kk, (long)Kk * H3d, (long)Kk * H3d, Bb, 0, 0,
                gb2, nullptr, 0, 1);

    // 11) GCN layer 3: H3 = relu(nadj @ (H2 @ w3) + b3)
    launch_gemm(stream, H2b, gw3, Tb, Kk, H4d, H3d, H3d, H4d, H4d,
                (long)Kk * H3d, 0, (long)Kk * H4d, Bb, 0, 0, nullptr, nullptr, 0, 0);
    launch_gemm(stream, ADJ, Tb, H3, Kk, H4d, Kk, Kk, H4d, H4d,
                (long)Kk * Kk, (long)Kk * H4d, (long)Kk * H4d, Bb, 0, 0,
                gb3, nullptr, 0, 1);

    // 12) residual head: OUTR = H3 @ adj_w + adj_b + NF
    launch_gemm(stream, H3, aw, OUTR, Kk, Hh, H4d, H4d, Hh, Hh,
                (long)Kk * H4d, 0, (long)Kk * Hh, Bb, 0, 0,
                ab, NF, (long)Kk * Hh, 0);

    // 13) FC1: X1 = gelu([2,49152] @ fc1_w + b)  (bandwidth floor: 196.6MB
    //     of weights; PREFETCH path streams fc1_w ahead via global_prefetch)
    launch_gemm(stream, OUTR, f1w, X1, Bb, FC1d, Kk * Hh, Kk * Hh, FC1d, FC1d,
                0, 0, 0, 1, 0, 1, f1b, nullptr, 0, 2);

    // 14) FC2: X2 = gelu(X1 @ fc2_w + b)
    launch_gemm(stream, X1, f2w, X2, Bb, FC2d, FC1d, FC1d, FC2d, FC2d,
                0, 0, 0, 1, 0, 0, f2b, nullptr, 0, 2);

    // 15) FC3 -> d_out [2,7]
    launch_gemm(stream, X2, f3w, out, Bb, NCd, FC2d, FC2d, NCd, NCd,
                0, 0, 0, 1, 0, 0, f3b, nullptr, 0, 0);

    (void)in_sizes; (void)n_in; (void)out_size; (void)ws_size;
}